// GatNet_2310692405647
// MI455X (gfx1250) — compile-verified
//
#include <hip/hip_runtime.h>
#include <math.h>

#define F_IN 128
#define H1   8
#define C1   32
#define HC1  256      // H1*C1
#define C2   40
#define NEG_SLOPE 0.2f

typedef float v2f __attribute__((ext_vector_type(2)));
typedef float v8f __attribute__((ext_vector_type(8)));

// ---------- order-preserving float <-> uint for atomicMax ----------
__device__ __forceinline__ unsigned f2ord(float f) {
    unsigned u = __float_as_uint(f);
    return (u & 0x80000000u) ? ~u : (u | 0x80000000u);
}
__device__ __forceinline__ float ord2f(unsigned k) {
    return (k & 0x80000000u) ? __uint_as_float(k & 0x7FFFFFFFu)
                             : __uint_as_float(~k);
}
#define ORD_NEG_INF 0x007FFFFFu   // f2ord(-inf)

// ---------- edge (src,dst) with appended self-loops ----------
__device__ __forceinline__ void edge_sd(const long long* __restrict__ ei,
                                        int e, int E, int& s, int& d) {
    if (e < E) { s = (int)ei[e]; d = (int)ei[E + e]; }
    else       { s = d = e - E; }
}

// =====================================================================
// Generic fp32 WMMA GEMM:  C[M,Nc] = A[M,K] @ B[K,Nc]   (row-major)
// one wave per 16x16 output tile; V_WMMA_F32_16X16X4_F32, K step 4.
// Branch-free inner loop: out-of-range columns are clamped for loads
// (their accumulator lanes are garbage but never stored), so EXEC
// stays all-1s and no exec-mask save/restore appears in the hot loop.
// =====================================================================
__global__ void gemm_wmma_f32(const float* __restrict__ A,
                              const float* __restrict__ B,
                              float* __restrict__ C,
                              int M, int K, int Nc, int nTiles) {
    int wave = blockIdx.x * (blockDim.x >> 5) + (threadIdx.x >> 5);
    int mTiles = (M + 15) >> 4;
    if (wave >= mTiles * nTiles) return;          // wave-uniform exit
    int lane = threadIdx.x & 31;
    int mT = wave / nTiles, nT = wave % nTiles;
    int half = lane >> 4;        // 0: lanes 0-15, 1: lanes 16-31
    int l16  = lane & 15;

    int row  = (mT << 4) + l16;                   // A row for this lane
    int rowc = row < M ? row : M - 1;             // clamp loads
    int col  = (nT << 4) + l16;                   // B/C column
    int colL = col < Nc ? col : Nc - 1;           // clamp loads

    const float* Ap = A + (size_t)rowc * K + (half << 1); // K=2*half,2*half+1
    const float* Bp = B + (size_t)(half << 1) * Nc + colL;
    const size_t bStep = (size_t)Nc * 4;

    v8f acc = {};
    #pragma unroll 8
    for (int k0 = 0; k0 < K; k0 += 4) {
        v2f a = *(const v2f*)Ap;                  // one b64 load
        v2f b; b.x = Bp[0]; b.y = Bp[Nc];         // rows k, k+1 of column
        acc = __builtin_amdgcn_wmma_f32_16x16x4_f32(
                  false, a, false, b, (short)0, acc, false, false);
        Ap += 4;
        Bp += bStep;
    }
    if (col < Nc) {
        int mBase = (mT << 4) + (half << 3);      // rows r and r+8 per half
        #pragma unroll
        for (int r = 0; r < 8; ++r) {
            int m = mBase + r;
            if (m < M) C[(size_t)m * Nc + col] = acc[r];
        }
    }
}

// =====================================================================
// init: zero accumulators, set segment-max buffers to -inf (ordered)
// =====================================================================
__global__ void init_kernel(float* __restrict__ z, unsigned* __restrict__ m1,
                            float* __restrict__ den1, unsigned* __restrict__ m2,
                            float* __restrict__ den2, float* __restrict__ agg2,
                            int N) {
    int i = blockIdx.x * blockDim.x + threadIdx.x;
    if (i < N * HC1) z[i] = 0.f;
    if (i < N * H1) { m1[i] = ORD_NEG_INF; den1[i] = 0.f; }
    if (i < N)      { m2[i] = ORD_NEG_INF; den2[i] = 0.f; }
    if (i < N * C2) agg2[i] = 0.f;
}

// =====================================================================
// attention coefficients
// =====================================================================
__global__ void att1_kernel(const float* __restrict__ h1,
                            const float* __restrict__ as1,
                            const float* __restrict__ ad1,
                            float* __restrict__ als, float* __restrict__ ald,
                            int N) {
    int idx = blockIdx.x * blockDim.x + threadIdx.x;
    if (idx >= N * H1) return;
    int n = idx / H1, h = idx % H1;
    const float* hp = h1 + (size_t)n * HC1 + h * C1;
    const float* sp = as1 + h * C1;
    const float* dp = ad1 + h * C1;
    float ss = 0.f, dd = 0.f;
    #pragma unroll
    for (int c = 0; c < C1; ++c) { float v = hp[c]; ss += v * sp[c]; dd += v * dp[c]; }
    als[idx] = ss; ald[idx] = dd;
}

__global__ void att2_kernel(const float* __restrict__ h2,
                            const float* __restrict__ as2,
                            const float* __restrict__ ad2,
                            float* __restrict__ als, float* __restrict__ ald,
                            int N) {
    int n = blockIdx.x * blockDim.x + threadIdx.x;
    if (n >= N) return;
    const float* hp = h2 + (size_t)n * C2;
    float ss = 0.f, dd = 0.f;
    #pragma unroll
    for (int c = 0; c < C2; ++c) { float v = hp[c]; ss += v * as2[c]; dd += v * ad2[c]; }
    als[n] = ss; ald[n] = dd;
}

// =====================================================================
// layer-1 edge passes (8 heads per edge)
// =====================================================================
__global__ void edge_max1(const long long* __restrict__ ei,
                          const float* __restrict__ als, const float* __restrict__ ald,
                          unsigned* __restrict__ m1, int E, int N) {
    int e = blockIdx.x * blockDim.x + threadIdx.x;
    if (e >= E + N) return;
    int s, d; edge_sd(ei, e, E, s, d);
    #pragma unroll
    for (int h = 0; h < H1; ++h) {
        float ev = als[s * H1 + h] + ald[d * H1 + h];
        ev = ev > 0.f ? ev : NEG_SLOPE * ev;
        atomicMax(&m1[d * H1 + h], f2ord(ev));
    }
}

__global__ void edge_den1(const long long* __restrict__ ei,
                          const float* __restrict__ als, const float* __restrict__ ald,
                          const unsigned* __restrict__ m1, float* __restrict__ den1,
                          int E, int N) {
    int e = blockIdx.x * blockDim.x + threadIdx.x;
    if (e >= E + N) return;
    int s, d; edge_sd(ei, e, E, s, d);
    #pragma unroll
    for (int h = 0; h < H1; ++h) {
        float ev = als[s * H1 + h] + ald[d * H1 + h];
        ev = ev > 0.f ? ev : NEG_SLOPE * ev;
        float w = expf(ev - ord2f(m1[d * H1 + h]));
        atomicAdd(&den1[d * H1 + h], w);
    }
}

__global__ void edge_alpha1(const long long* __restrict__ ei,
                            const float* __restrict__ als, const float* __restrict__ ald,
                            const unsigned* __restrict__ m1, const float* __restrict__ den1,
                            float* __restrict__ aE, int E, int N) {
    int e = blockIdx.x * blockDim.x + threadIdx.x;
    if (e >= E + N) return;
    int s, d; edge_sd(ei, e, E, s, d);
    #pragma unroll
    for (int h = 0; h < H1; ++h) {
        float ev = als[s * H1 + h] + ald[d * H1 + h];
        ev = ev > 0.f ? ev : NEG_SLOPE * ev;
        float w = expf(ev - ord2f(m1[d * H1 + h]));
        aE[e * H1 + h] = w / (den1[d * H1 + h] + 1e-16f);
    }
}

__global__ void edge_scat1(const long long* __restrict__ ei,
                           const float* __restrict__ h1,
                           const float* __restrict__ aE,
                           float* __restrict__ z, int E, int N) {
    long long idx = (long long)blockIdx.x * blockDim.x + threadIdx.x;
    if (idx >= (long long)(E + N) * HC1) return;
    int e = (int)(idx >> 8);
    int c = (int)(idx & 255);
    int s, d; edge_sd(ei, e, E, s, d);
    float a = aE[e * H1 + (c >> 5)];
    atomicAdd(&z[(size_t)d * HC1 + c], h1[(size_t)s * HC1 + c] * a);
}

__global__ void fin1_kernel(float* __restrict__ z, const float* __restrict__ b1, int N) {
    int idx = blockIdx.x * blockDim.x + threadIdx.x;
    if (idx >= N * HC1) return;
    float v = z[idx] + b1[idx & 255];
    z[idx] = v > 0.f ? v : expm1f(v);                 // ELU
}

// =====================================================================
// layer-2 edge passes (1 head, 40 channels)
// =====================================================================
__global__ void edge_max2(const long long* __restrict__ ei,
                          const float* __restrict__ als, const float* __restrict__ ald,
                          unsigned* __restrict__ m2, int E, int N) {
    int e = blockIdx.x * blockDim.x + threadIdx.x;
    if (e >= E + N) return;
    int s, d; edge_sd(ei, e, E, s, d);
    float ev = als[s] + ald[d];
    ev = ev > 0.f ? ev : NEG_SLOPE * ev;
    atomicMax(&m2[d], f2ord(ev));
}

__global__ void edge_den2(const long long* __restrict__ ei,
                          const float* __restrict__ als, const float* __restrict__ ald,
                          const unsigned* __restrict__ m2, float* __restrict__ den2,
                          int E, int N) {
    int e = blockIdx.x * blockDim.x + threadIdx.x;
    if (e >= E + N) return;
    int s, d; edge_sd(ei, e, E, s, d);
    float ev = als[s] + ald[d];
    ev = ev > 0.f ? ev : NEG_SLOPE * ev;
    atomicAdd(&den2[d], expf(ev - ord2f(m2[d])));
}

__global__ void edge_alpha2(const long long* __restrict__ ei,
                            const float* __restrict__ als, const float* __restrict__ ald,
                            const unsigned* __restrict__ m2, const float* __restrict__ den2,
                            float* __restrict__ aE, int E, int N) {
    int e = blockIdx.x * blockDim.x + threadIdx.x;
    if (e >= E + N) return;
    int s, d; edge_sd(ei, e, E, s, d);
    float ev = als[s] + ald[d];
    ev = ev > 0.f ? ev : NEG_SLOPE * ev;
    float w = expf(ev - ord2f(m2[d]));
    aE[e] = w / (den2[d] + 1e-16f);
}

__global__ void edge_scat2(const long long* __restrict__ ei,
                           const float* __restrict__ h2,
                           const float* __restrict__ aE,
                           float* __restrict__ agg2, int E, int N) {
    long long idx = (long long)blockIdx.x * blockDim.x + threadIdx.x;
    if (idx >= (long long)(E + N) * C2) return;
    int e = (int)(idx / C2);
    int c = (int)(idx - (long long)e * C2);
    int s, d; edge_sd(ei, e, E, s, d);
    atomicAdd(&agg2[(size_t)d * C2 + c], h2[(size_t)s * C2 + c] * aE[e]);
}

__global__ void fin2_kernel(const float* __restrict__ agg2,
                            const float* __restrict__ b2,
                            float* __restrict__ out, int N) {
    int n = blockIdx.x * blockDim.x + threadIdx.x;
    if (n >= N) return;
    float v[C2];
    float mx = -INFINITY;
    #pragma unroll
    for (int c = 0; c < C2; ++c) { v[c] = agg2[(size_t)n * C2 + c] + b2[c]; mx = fmaxf(mx, v[c]); }
    float ssum = 0.f;
    #pragma unroll
    for (int c = 0; c < C2; ++c) { float ex = expf(v[c] - mx); v[c] = ex; ssum += ex; }
    float inv = 1.f / ssum;
    #pragma unroll
    for (int c = 0; c < C2; ++c) out[(size_t)n * C2 + c] = v[c] * inv;
}

// =====================================================================
extern "C" void kernel_launch(void* const* d_in, const int* in_sizes, int n_in,
                              void* d_out, int out_size, void* d_ws, size_t ws_size,
                              hipStream_t stream) {
    const float*     x   = (const float*)d_in[0];
    const long long* ei  = (const long long*)d_in[1];   // int64 edge_index [2,E]
    const float*     W1  = (const float*)d_in[2];
    const float*     as1 = (const float*)d_in[3];
    const float*     ad1 = (const float*)d_in[4];
    const float*     b1  = (const float*)d_in[5];
    const float*     W2  = (const float*)d_in[6];
    const float*     as2 = (const float*)d_in[7];
    const float*     ad2 = (const float*)d_in[8];
    const float*     b2  = (const float*)d_in[9];

    const int N  = in_sizes[0] / F_IN;
    const int E  = in_sizes[1] / 2;
    const int Et = E + N;

    float* z   = (float*)d_out;                       // [N, 256] (output 1)
    float* out = (float*)d_out + (size_t)N * HC1;     // [N, 40]  (output 2)

    float* ws = (float*)d_ws;
    size_t off = 0;
    float*    h1   = ws + off; off += (size_t)N * HC1;
    float*    als1 = ws + off; off += (size_t)N * H1;
    float*    ald1 = ws + off; off += (size_t)N * H1;
    unsigned* m1   = (unsigned*)(ws + off); off += (size_t)N * H1;
    float*    den1 = ws + off; off += (size_t)N * H1;
    float*    aE1  = ws + off; off += (size_t)Et * H1;
    float*    h2   = ws + off; off += (size_t)N * C2;
    float*    als2 = ws + off; off += (size_t)N;
    float*    ald2 = ws + off; off += (size_t)N;
    unsigned* m2   = (unsigned*)(ws + off); off += (size_t)N;
    float*    den2 = ws + off; off += (size_t)N;
    float*    aE2  = ws + off; off += (size_t)Et;
    float*    agg2 = ws + off; off += (size_t)N * C2;

    const int TB = 256;
    #define NB(n) ((int)(((long long)(n) + TB - 1) / TB))

    init_kernel<<<NB((size_t)N * HC1), TB, 0, stream>>>(z, m1, den1, m2, den2, agg2, N);

    { // layer-1 GEMM: h1 = x @ W1   (M=N, K=128, Nc=256)
        int nT = HC1 / 16, tiles = ((N + 15) / 16) * nT;
        gemm_wmma_f32<<<(tiles + 7) / 8, 256, 0, stream>>>(x, W1, h1, N, F_IN, HC1, nT);
    }
    att1_kernel<<<NB(N * H1), TB, 0, stream>>>(h1, as1, ad1, als1, ald1, N);
    edge_max1  <<<NB(Et), TB, 0, stream>>>(ei, als1, ald1, m1, E, N);
    edge_den1  <<<NB(Et), TB, 0, stream>>>(ei, als1, ald1, m1, den1, E, N);
    edge_alpha1<<<NB(Et), TB, 0, stream>>>(ei, als1, ald1, m1, den1, aE1, E, N);
    edge_scat1 <<<NB((long long)Et * HC1), TB, 0, stream>>>(ei, h1, aE1, z, E, N);
    fin1_kernel<<<NB((size_t)N * HC1), TB, 0, stream>>>(z, b1, N);

    { // layer-2 GEMM: h2 = z @ W2   (M=N, K=256, Nc=40 -> 3 col tiles)
        int nT = (C2 + 15) / 16, tiles = ((N + 15) / 16) * nT;
        gemm_wmma_f32<<<(tiles + 7) / 8, 256, 0, stream>>>(z, W2, h2, N, HC1, C2, nT);
    }
    att2_kernel<<<NB(N), TB, 0, stream>>>(h2, as2, ad2, als2, ald2, N);
    edge_max2  <<<NB(Et), TB, 0, stream>>>(ei, als2, ald2, m2, E, N);
    edge_den2  <<<NB(Et), TB, 0, stream>>>(ei, als2, ald2, m2, den2, E, N);
    edge_alpha2<<<NB(Et), TB, 0, stream>>>(ei, als2, ald2, m2, den2, aE2, E, N);
    edge_scat2 <<<NB((long long)Et * C2), TB, 0, stream>>>(ei, h2, aE2, agg2, E, N);
    fin2_kernel<<<NB(N), TB, 0, stream>>>(agg2, b2, out, N);

    #undef NB
    (void)n_in; (void)out_size; (void)ws_size;
}